// RGCNLayer_39393440039445
// MI455X (gfx1250) — compile-verified
//
#include <hip/hip_runtime.h>
#include <hip/hip_bf16.h>

// Problem constants (match reference).
#define NN    50000      // nodes
#define RR    8          // relations
#define H0    128        // in features
#define H1    64         // out features
#define NB    16         // bases

typedef __attribute__((ext_vector_type(2))) float v2f;
typedef __attribute__((ext_vector_type(8))) float v8f;

// ---------------------------------------------------------------------------
// 1) W[r,i,o] = sum_b comps[r,b] * bases[b,i,o]    (tiny: 64K outputs)
// ---------------------------------------------------------------------------
__global__ void basis_weights(const float* __restrict__ comps,
                              const float* __restrict__ bases,
                              float* __restrict__ W) {
    int idx = blockIdx.x * blockDim.x + threadIdx.x;
    if (idx >= RR * H0 * H1) return;
    int r  = idx / (H0 * H1);
    int io = idx % (H0 * H1);
    float acc = 0.0f;
#pragma unroll
    for (int b = 0; b < NB; ++b)
        acc = fmaf(comps[r * NB + b], bases[b * H0 * H1 + io], acc);
    W[idx] = acc;
}

// ---------------------------------------------------------------------------
// 2) colsum[c] += 1 for every edge (column-degree normalization)
// ---------------------------------------------------------------------------
__global__ void col_count(const int* __restrict__ cols,
                          float* __restrict__ colsum, int nnz) {
    int e = blockIdx.x * blockDim.x + threadIdx.x;
    if (e < nnz) atomicAdd(&colsum[cols[e]], 1.0f);
}

// ---------------------------------------------------------------------------
// 3) out[n,o] = bias[o]  (SpMM atomics then accumulate on top of the bias)
// ---------------------------------------------------------------------------
__global__ void init_out(float* __restrict__ out, const float* __restrict__ bias) {
    int idx = blockIdx.x * blockDim.x + threadIdx.x;
    if (idx < NN * H1) out[idx] = bias[idx & (H1 - 1)];
}

// ---------------------------------------------------------------------------
// 4) nw[r*N + n, o] = sum_i nodes[n,i] * W[r,i,o]
//    Dense GEMM via V_WMMA_F32_16X16X4_F32, one wave32 per 16x16 tile.
//    50000 = 3125 * 16 exactly, H1 = 4 * 16 exactly -> no edge masking,
//    EXEC stays all-ones as WMMA requires.
// ---------------------------------------------------------------------------
__global__ void gemm_nw(const float* __restrict__ nodes,
                        const float* __restrict__ W,
                        float* __restrict__ nw) {
    const int MT = NN / 16;   // 3125 M-tiles
    const int NT = H1 / 16;   // 4 N-tiles

    int wave = (blockIdx.x * blockDim.x + threadIdx.x) >> 5;
    int lane = threadIdx.x & 31;
    if (wave >= RR * MT * NT) return;

    int nt  = wave % NT;
    int t   = wave / NT;
    int mt  = t % MT;
    int r   = t / MT;

    const float* __restrict__ A = nodes + (size_t)(mt * 16) * H0;      // 16 x 128
    const float* __restrict__ B = W + (size_t)r * H0 * H1 + nt * 16;   // 128 x 64, col slice
    float* __restrict__ C = nw + ((size_t)(r * NN + mt * 16)) * H1 + nt * 16;

    const int mrow  = lane & 15;          // row within tile (A,M) / col (B,N / C,N)
    const int khalf = (lane >> 4) * 2;    // K sub-offset: lanes 0-15 -> {0,1}, 16-31 -> {2,3}

    v8f acc = {};
#pragma unroll
    for (int k0 = 0; k0 < H0; k0 += 4) {
        v2f a, b;
        // A 16x4 f32 fragment: VGPR0 = K=k0+khalf, VGPR1 = K=k0+khalf+1 (ISA 7.12.2)
        a.x = A[(size_t)mrow * H0 + k0 + khalf + 0];
        a.y = A[(size_t)mrow * H0 + k0 + khalf + 1];
        // B 4x16 f32 fragment, mirrored striping: row K across half-waves, N = lane%16
        b.x = B[(size_t)(k0 + khalf + 0) * H1 + mrow];
        b.y = B[(size_t)(k0 + khalf + 1) * H1 + mrow];
        acc = __builtin_amdgcn_wmma_f32_16x16x4_f32(
            /*neg_a=*/false, a, /*neg_b=*/false, b,
            /*c_mod=*/(short)0, acc, /*reuse_a=*/false, /*reuse_b=*/false);
    }

    // C/D layout: VGPR i -> row (i + 8*(lane/16)), col lane%16
    const int rbase = (lane >> 4) * 8;
#pragma unroll
    for (int i = 0; i < 8; ++i)
        C[(size_t)(rbase + i) * H1 + mrow] = acc[i];
}

// ---------------------------------------------------------------------------
// 5) SpMM scatter: out[row] += (1/colsum[col]) * nw[col]
//    One wave per edge; lane covers outputs {lane, lane+32}. Coalesced 256B
//    row gathers from nw (mostly L2-resident: nw=102MB vs 192MB L2).
// ---------------------------------------------------------------------------
__global__ void spmm(const int* __restrict__ rows,
                     const int* __restrict__ cols,
                     const float* __restrict__ colsum,
                     const float* __restrict__ nw,
                     float* __restrict__ out, int nnz) {
    long long t = (long long)blockIdx.x * blockDim.x + threadIdx.x;
    int e    = (int)(t >> 5);
    int lane = (int)(t & 31);
    if (e >= nnz) return;

    int c = cols[e];
    int rdst = rows[e];
    float inv = 1.0f / colsum[c];          // every column has >= 1 entry

    const float* __restrict__ src = nw + (size_t)c * H1;
    float v0 = src[lane]      * inv;
    float v1 = src[lane + 32] * inv;

    float* dst = out + (size_t)rdst * H1;
    atomicAdd(&dst[lane],      v0);
    atomicAdd(&dst[lane + 32], v1);
}

// ---------------------------------------------------------------------------
// launch
// ---------------------------------------------------------------------------
extern "C" void kernel_launch(void* const* d_in, const int* in_sizes, int n_in,
                              void* d_out, int out_size, void* d_ws, size_t ws_size,
                              hipStream_t stream) {
    const float* nodes = (const float*)d_in[0];   // [N, H0]
    const float* comps = (const float*)d_in[1];   // [R, NB]
    const float* bases = (const float*)d_in[2];   // [NB, H0, H1]
    const float* bias  = (const float*)d_in[3];   // [H1]
    const int*   rows  = (const int*)d_in[4];     // [NNZ]
    const int*   cols  = (const int*)d_in[5];     // [NNZ]
    float*       out   = (float*)d_out;           // [N, H1]
    const int    nnz   = in_sizes[4];

    // Workspace carve-up (256B aligned):
    //   W      : R*H0*H1 floats          = 262,144 B   @ 0
    //   colsum : R*N floats              = 1,600,000 B @ 262,144
    //   nw     : R*N*H1 floats           = 102,400,000 B @ 1,862,144
    char*  ws     = (char*)d_ws;
    float* W      = (float*)(ws);
    float* colsum = (float*)(ws + 262144);
    float* nw     = (float*)(ws + 1862144);

    hipMemsetAsync(colsum, 0, (size_t)RR * NN * sizeof(float), stream);

    {   // 1) basis -> per-relation weights
        int n = RR * H0 * H1;
        basis_weights<<<(n + 255) / 256, 256, 0, stream>>>(comps, bases, W);
    }
    {   // 2) column degree counts
        col_count<<<(nnz + 255) / 256, 256, 0, stream>>>(cols, colsum, nnz);
    }
    {   // 3) seed output with bias
        int n = NN * H1;
        init_out<<<(n + 255) / 256, 256, 0, stream>>>(out, bias);
    }
    {   // 4) WMMA GEMM: nw = nodes @ W_r  for all relations
        int waves  = RR * (NN / 16) * (H1 / 16);    // 100,000 waves
        int blocks = (waves * 32 + 255) / 256;      // 8 waves / 256-thread block
        gemm_nw<<<blocks, 256, 0, stream>>>(nodes, W, nw);
    }
    {   // 5) SpMM scatter with degree normalization
        long long threads = (long long)nnz * 32;
        int blocks = (int)((threads + 255) / 256);
        spmm<<<blocks, 256, 0, stream>>>(rows, cols, colsum, nw, out, nnz);
    }
}